// OvOAttention_6055903888044
// MI455X (gfx1250) — compile-verified
//
#include <hip/hip_runtime.h>
#include <math.h>

// ---- problem constants (match reference: B=16384, H=16, S=1, E=64) ----
#define BATCH 16384
#define H 16
#define E 64
#define WAVES_PER_BLOCK 4
#define THREADS (WAVES_PER_BLOCK * 32)

typedef __attribute__((ext_vector_type(2))) float v2f;
typedef __attribute__((ext_vector_type(8))) float v8f;

// D = A(16x4,f32) * B(4x16,f32) + C(16x16,f32)  -> v_wmma_f32_16x16x4_f32
#define WMMA4(accv, av, bv)                                                    \
    accv = __builtin_amdgcn_wmma_f32_16x16x4_f32(                              \
        /*neg_a=*/false, (av), /*neg_b=*/false, (bv),                          \
        /*c_mod=*/(short)0, (accv), /*reuse_a=*/false, /*reuse_b=*/false)

__global__ __launch_bounds__(THREADS) void ovo_attn_kernel(
    const float* __restrict__ o0, const float* __restrict__ o1,
    const float* __restrict__ o2, const float* __restrict__ o3,
    const float* __restrict__ mainp, const float* __restrict__ W,
    float* __restrict__ ctx_out, float* __restrict__ attn_out)
{
    // LDS: W shared by block; per-wave staging for m/t, main, score/attn.
    __shared__ __align__(16) float sW[E * E];                      // 16 KB
    __shared__ __align__(16) float sM[WAVES_PER_BLOCK][H * E];     // 4 KB/wave (m, then t)
    __shared__ __align__(16) float sK[WAVES_PER_BLOCK][H * E];     // 4 KB/wave (main)
    __shared__ __align__(16) float sA[WAVES_PER_BLOCK][H * H];     // 1 KB/wave (score/attn)

    const int tid  = threadIdx.x;
    const int w    = tid >> 5;        // wave in block
    const int lane = tid & 31;
    const int ln   = lane & 15;       // M (A-frag row) == N (B/C/D col) index
    const int kh   = lane >> 4;       // K half-select for A/B fragments
    const int b    = blockIdx.x * WAVES_PER_BLOCK + w;

    // ---- stage W into LDS (block-cooperative, float4) ----
    {
        const float4* W4  = (const float4*)W;
        float4*       sW4 = (float4*)sW;
        #pragma unroll
        for (int t = 0; t < (E * E / 4) / THREADS; ++t)
            sW4[tid + t * THREADS] = W4[tid + t * THREADS];
    }

    // ---- stage mean(others) and main for this wave's batch element ----
    {
        const size_t  base4 = (size_t)b * (H * E / 4);
        const float4* a4 = (const float4*)o0 + base4;
        const float4* b4 = (const float4*)o1 + base4;
        const float4* c4 = (const float4*)o2 + base4;
        const float4* d4 = (const float4*)o3 + base4;
        const float4* m4 = (const float4*)mainp + base4;
        float4* sM4 = (float4*)sM[w];
        float4* sK4 = (float4*)sK[w];
        #pragma unroll
        for (int t = 0; t < (H * E / 4) / 32; ++t) {
            const int i = lane + t * 32;
            float4 x0 = a4[i], x1 = b4[i], x2 = c4[i], x3 = d4[i];
            float4 r;
            r.x = (x0.x + x1.x + x2.x + x3.x) * 0.25f;
            r.y = (x0.y + x1.y + x2.y + x3.y) * 0.25f;
            r.z = (x0.z + x1.z + x2.z + x3.z) * 0.25f;
            r.w = (x0.w + x1.w + x2.w + x3.w) * 0.25f;
            sM4[i] = r;
            sK4[i] = m4[i];
        }
    }
    __syncthreads();

    // ---- GEMM1: t = m @ W   (16x64)·(64x64) -> 4 D-tiles of 16x16 ----
    v8f acc0 = {}, acc1 = {}, acc2 = {}, acc3 = {};
    #pragma unroll
    for (int k0 = 0; k0 < E; k0 += 4) {
        const int ka = k0 + 2 * kh;
        v2f a;
        a.x = sM[w][ln * E + ka + 0];
        a.y = sM[w][ln * E + ka + 1];
        v2f b0, b1, b2, b3;
        b0.x = sW[(ka + 0) * E +  0 + ln];  b0.y = sW[(ka + 1) * E +  0 + ln];
        b1.x = sW[(ka + 0) * E + 16 + ln];  b1.y = sW[(ka + 1) * E + 16 + ln];
        b2.x = sW[(ka + 0) * E + 32 + ln];  b2.y = sW[(ka + 1) * E + 32 + ln];
        b3.x = sW[(ka + 0) * E + 48 + ln];  b3.y = sW[(ka + 1) * E + 48 + ln];
        WMMA4(acc0, a, b0);
        WMMA4(acc1, a, b1);
        WMMA4(acc2, a, b2);
        WMMA4(acc3, a, b3);
    }

    // spill t (D-layout) -> LDS row-major, overwriting m (wave-private buffer)
    #pragma unroll
    for (int v = 0; v < 8; ++v) {
        const int m = v + 8 * kh;
        sM[w][m * E +  0 + ln] = acc0[v];
        sM[w][m * E + 16 + ln] = acc1[v];
        sM[w][m * E + 32 + ln] = acc2[v];
        sM[w][m * E + 48 + ln] = acc3[v];
    }
    __syncthreads();

    // ---- GEMM2: score = t @ mainᵀ   (16x64)·(64x16) -> one 16x16 tile ----
    v8f s = {};
    #pragma unroll
    for (int k0 = 0; k0 < E; k0 += 4) {
        const int ka = k0 + 2 * kh;
        v2f a, bb;
        a.x  = sM[w][ln * E + ka + 0];
        a.y  = sM[w][ln * E + ka + 1];
        bb.x = sK[w][ln * E + ka + 0];   // B[k][n] = main[n][k]
        bb.y = sK[w][ln * E + ka + 1];
        WMMA4(s, a, bb);
    }
    #pragma unroll
    for (int v = 0; v < 8; ++v)
        sA[w][(v + 8 * kh) * H + ln] = s[v];
    __syncthreads();

    // ---- softmax over rows of score (16 rows, one lane each) ----
    if (lane < 16) {
        float vals[H];
        float mx = -INFINITY;
        #pragma unroll
        for (int j = 0; j < H; ++j) { vals[j] = sA[w][lane * H + j]; mx = fmaxf(mx, vals[j]); }
        float sum = 0.0f;
        #pragma unroll
        for (int j = 0; j < H; ++j) { vals[j] = expf(vals[j] - mx); sum += vals[j]; }
        const float inv = 1.0f / sum;
        #pragma unroll
        for (int j = 0; j < H; ++j) {
            const float p = vals[j] * inv;
            sA[w][lane * H + j] = p;
            attn_out[(size_t)b * (H * H) + lane * H + j] = p;
        }
    }
    __syncthreads();

    // ---- GEMM3: context = attn @ main   (16x16)·(16x64) -> 4 D-tiles ----
    v8f c0 = {}, c1 = {}, c2 = {}, c3 = {};
    #pragma unroll
    for (int k0 = 0; k0 < H; k0 += 4) {
        const int ka = k0 + 2 * kh;
        v2f a;
        a.x = sA[w][ln * H + ka + 0];
        a.y = sA[w][ln * H + ka + 1];
        v2f b0, b1, b2, b3;
        b0.x = sK[w][(ka + 0) * E +  0 + ln];  b0.y = sK[w][(ka + 1) * E +  0 + ln];
        b1.x = sK[w][(ka + 0) * E + 16 + ln];  b1.y = sK[w][(ka + 1) * E + 16 + ln];
        b2.x = sK[w][(ka + 0) * E + 32 + ln];  b2.y = sK[w][(ka + 1) * E + 32 + ln];
        b3.x = sK[w][(ka + 0) * E + 48 + ln];  b3.y = sK[w][(ka + 1) * E + 48 + ln];
        WMMA4(c0, a, b0);
        WMMA4(c1, a, b1);
        WMMA4(c2, a, b2);
        WMMA4(c3, a, b3);
    }

    // ---- store context (D-layout -> row-major global) ----
    float* ctxb = ctx_out + (size_t)b * (H * E);
    #pragma unroll
    for (int v = 0; v < 8; ++v) {
        const int m = v + 8 * kh;
        ctxb[m * E +  0 + ln] = c0[v];
        ctxb[m * E + 16 + ln] = c1[v];
        ctxb[m * E + 32 + ln] = c2[v];
        ctxb[m * E + 48 + ln] = c3[v];
    }
}

extern "C" void kernel_launch(void* const* d_in, const int* in_sizes, int n_in,
                              void* d_out, int out_size, void* d_ws, size_t ws_size,
                              hipStream_t stream) {
    (void)in_sizes; (void)n_in; (void)out_size; (void)d_ws; (void)ws_size;
    const float* o0 = (const float*)d_in[0];
    const float* o1 = (const float*)d_in[1];
    const float* o2 = (const float*)d_in[2];
    const float* o3 = (const float*)d_in[3];
    const float* mp = (const float*)d_in[4];
    const float* W  = (const float*)d_in[5];

    float* ctx  = (float*)d_out;                       // (B,H,E) flat
    float* attn = ctx + (size_t)BATCH * H * E;         // (B,H,H) flat

    dim3 grid(BATCH / WAVES_PER_BLOCK);
    dim3 block(THREADS);
    ovo_attn_kernel<<<grid, block, 0, stream>>>(o0, o1, o2, o3, mp, W, ctx, attn);
}